// Model_54683523612735
// MI455X (gfx1250) — compile-verified
//
#include <hip/hip_runtime.h>
#include <hip/hip_bf16.h>
#include <stdint.h>

// Problem constants (from reference)
#define Bc   32
#define Tc   1024
#define U0c  128
#define Uc   257          // 2*U0+1
#define Vc   1024
#define ERc  132          // compact emission row stride: slot0=blank, 1..128=labels (padded)
#define BPRc 260          // backpointer row stride (bytes, u8 deltas)
#define DLS  264          // dpLast row stride (floats)
#define CHK  64           // backtrace chunk (timesteps) staged in LDS
#define NEGF (-3.4028234663852886e38f)

// ---------------- CDNA5 async global->LDS helpers ----------------
__device__ __forceinline__ unsigned lds_off(const void* p) {
  // generic (flat) LDS address: low 32 bits are the LDS byte offset
  return (unsigned)(uintptr_t)p;
}
__device__ __forceinline__ void async_ld_b128(unsigned lds, unsigned voff, const void* sbase) {
  asm volatile("global_load_async_to_lds_b128 %0, %1, %2"
               :: "v"(lds), "v"(voff), "s"(sbase) : "memory");
}
__device__ __forceinline__ void async_wait0() {
  asm volatile("s_wait_asynccnt 0" ::: "memory");
}
__device__ __forceinline__ void async_wait1() {
  // ASYNCcnt completes in order: "<=1 outstanding" proves all but the newest load landed
  asm volatile("s_wait_asynccnt 1" ::: "memory");
}

// ---------------- wave32 butterfly reductions via ds_swizzle (1 ds op / stage) ----
template <int P>
__device__ __forceinline__ float swz_max(float v) {
  const int r = __builtin_amdgcn_ds_swizzle(__float_as_int(v), P);
  return fmaxf(v, __int_as_float(r));
}
template <int P>
__device__ __forceinline__ float swz_add(float v) {
  const int r = __builtin_amdgcn_ds_swizzle(__float_as_int(v), P);
  return v + __int_as_float(r);
}
__device__ __forceinline__ float wave_max(float v) {
  v = swz_max<0x401f>(v);  // xor 16
  v = swz_max<0x201f>(v);  // xor 8
  v = swz_max<0x101f>(v);  // xor 4
  v = swz_max<0x081f>(v);  // xor 2
  v = swz_max<0x041f>(v);  // xor 1
  return v;
}
__device__ __forceinline__ float wave_sum(float v) {
  v = swz_add<0x401f>(v);
  v = swz_add<0x201f>(v);
  v = swz_add<0x101f>(v);
  v = swz_add<0x081f>(v);
  v = swz_add<0x041f>(v);
  return v;
}

// ---------------- Kernel 1: log-softmax + compact emission gather ----------------
// grid = B*T blocks, 256 threads (8 waves). Row lives in LDS; gather hits LDS.
__global__ __launch_bounds__(256) void ctc_lsm(const float* __restrict__ logits,
                                               const int*   __restrict__ label,
                                               float*       __restrict__ emitC) {
  __shared__ __align__(16) float xs[Vc];
  __shared__ float redm[8];
  __shared__ float reds[8];
  const int tid = threadIdx.x;
  const int bt  = blockIdx.x;
  const int b   = bt >> 10;                       // T = 1024

  const float4 x = ((const float4*)(logits + (size_t)bt * Vc))[tid];
  ((float4*)xs)[tid] = x;

  float m = fmaxf(fmaxf(x.x, x.y), fmaxf(x.z, x.w));
  m = wave_max(m);
  if ((tid & 31) == 0) redm[tid >> 5] = m;
  __syncthreads();
  if (tid == 0) {
    float mm = redm[0];
    for (int i = 1; i < 8; ++i) mm = fmaxf(mm, redm[i]);
    redm[0] = mm;
  }
  __syncthreads();
  const float M = redm[0];

  float s = __expf(x.x - M) + __expf(x.y - M) + __expf(x.z - M) + __expf(x.w - M);
  s = wave_sum(s);
  if ((tid & 31) == 0) reds[tid >> 5] = s;
  __syncthreads();
  if (tid == 0) {
    float ss = 0.f;
    for (int i = 0; i < 8; ++i) ss += reds[i];
    reds[0] = M + __logf(ss);
  }
  __syncthreads();
  const float logZ = reds[0];

  // compact emissions: slot 0 = blank (index 0); slot 1+k = logp at label[b][k]
  // (label==0 means padding; reference clips it to index 0 == blank anyway)
  if (tid < 1 + U0c) {
    const int v = (tid == 0) ? 0 : label[b * U0c + tid - 1];
    emitC[(size_t)bt * ERc + tid] = xs[v] - logZ;
  }
}

// ---------------- Kernel 2: sequential DP over T, triple-buffered async emissions ----
// grid = B blocks (one WGP each), 320 threads (10 waves); thread u handles lattice cell u.
// Step t: issue async row t+2, compute from row t (resident), wait asynccnt<=1 (row t+1
// proven resident by in-order completion), barrier. One full step of latency hiding.
__global__ __launch_bounds__(320) void ctc_dp(const float* __restrict__ emitC,
                                              const int*   __restrict__ label,
                                              unsigned char* __restrict__ bp,
                                              float*       __restrict__ dpLast) {
  __shared__ __align__(16) float em[3][ERc];       // emission row triple buffer (528B each)
  __shared__ __align__(16) float dpbuf[2][Uc + 7]; // dp[u] at slot u+2; slots 0,1 = NEG guards
  const int tid = threadIdx.x;
  const int b   = blockIdx.x;
  const int u   = tid;

  // per-cell static info: emission slot and "cond" (blank or repeat-2-back)
  int  eidx = 0;
  bool cond = true;
  if (u < Uc && (u & 1)) {
    const int k = u >> 1;
    eidx = 1 + k;
    const int lk = label[b * U0c + k];
    cond = (k == 0) ? (lk == 0) : (lk == label[b * U0c + k - 1]);
  }

  if (tid < Uc + 7) { dpbuf[0][tid] = NEGF; dpbuf[1][tid] = NEGF; }

  // row 0 sync; rows 1,2 async into em[1], em[2]
  const float* row0 = emitC + (size_t)(b * Tc) * ERc;
  if (tid < ERc) em[0][tid] = row0[tid];
  if (tid < ERc / 4) {
    async_ld_b128(lds_off(&em[1][tid * 4]), tid * 16,
                  emitC + (size_t)(b * Tc + 1) * ERc);
    async_ld_b128(lds_off(&em[2][tid * 4]), tid * 16,
                  emitC + (size_t)(b * Tc + 2) * ERc);
  }
  async_wait1();                 // row 1 resident (oldest of the two)
  __syncthreads();
  if (tid == 0) dpbuf[0][2] = em[0][0];   // dp0[0] = blank emission
  if (tid == 1) dpbuf[0][3] = em[0][1];   // dp0[1] = first label emission
  __syncthreads();

  unsigned char* bpb = bp + (size_t)b * Tc * BPRc;

  int ecur = 1;                  // t % 3 for t = 1
  for (int t = 1; t < Tc; ++t) {
    const int cur = t & 1, prv = cur ^ 1;
    const bool issue = (t + 2 < Tc);
    if (issue && tid < ERc / 4) {
      const int enx = (ecur + 2 >= 3) ? (ecur - 1) : (ecur + 2);   // (t+2) % 3
      async_ld_b128(lds_off(&em[enx][tid * 4]), tid * 16,
                    emitC + (size_t)(b * Tc + t + 2) * ERc);
    }
    if (u < Uc) {
      const float* rb = dpbuf[prv];
      const float d0 = rb[u + 2], d1 = rb[u + 1], d2 = rb[u];
      const float e  = em[ecur][eidx];
      float a; int delta;
      if (cond) {
        const float m = fmaxf(d0, d1);
        a = m + __logf(__expf(d0 - m) + __expf(d1 - m));
        delta = (d0 >= d1) ? 0 : 1;                       // argmax, first max wins
      } else {
        const float m = fmaxf(fmaxf(d0, d1), d2);
        a = m + __logf(__expf(d0 - m) + __expf(d1 - m) + __expf(d2 - m));
        delta = (d0 >= d1 && d0 >= d2) ? 0 : ((d1 >= d2) ? 1 : 2);
      }
      dpbuf[cur][u + 2] = e + a;
      bpb[(size_t)t * BPRc + u] = (unsigned char)delta;   // bp[t][u] = u - delta
    }
    if (issue) async_wait1();    // row t+1 resident; row t+2 still in flight
    else       async_wait0();    // tail: everything must be resident
    __syncthreads();
    ecur = (ecur == 2) ? 0 : ecur + 1;
  }
  if (u < Uc) dpLast[(size_t)b * DLS + u] = dpbuf[(Tc - 1) & 1][u + 2];
}

// ---------------- Kernel 3: loss + backward traversal (LDS-staged, chunked) ----------------
// grid = B blocks. Sequential 1024-step pointer chase at LDS latency; next chunk of
// bp deltas streams in via async loads while lane 0 walks the current chunk.
__global__ __launch_bounds__(256) void ctc_back(const unsigned char* __restrict__ bp,
                                                const float* __restrict__ dpLast,
                                                const int*   __restrict__ label,
                                                int*         __restrict__ wis,
                                                float*       __restrict__ outLoss) {
  __shared__ __align__(16) unsigned char bpl[2][CHK * BPRc]; // 2 x 16640 B
  __shared__ int sh_cnt;
  __shared__ int sh_wi;
  const int tid = threadIdx.x;
  const int b   = blockIdx.x;

  if (tid == 0) sh_cnt = 0;
  __syncthreads();
  if (tid < U0c && label[b * U0c + tid] != 0) atomicAdd(&sh_cnt, 1);
  __syncthreads();
  if (tid == 0) {
    const int yl = 2 * sh_cnt + 1;
    const int i1 = yl - 1;
    int i2 = yl - 2; if (i2 < 0) i2 += Uc;   // emulate jnp negative-index wrap
    const float d1 = dpLast[(size_t)b * DLS + i1];
    const float d2 = dpLast[(size_t)b * DLS + i2];
    const float m  = fmaxf(d1, d2);
    const float l  = m + __logf(__expf(d1 - m) + __expf(d2 - m));
    outLoss[b] = (-l) * 2.0f / (float)(yl - 1);
    sh_wi = (d1 > d2) ? i1 : i2;
  }

  const unsigned char* bpb = bp + (size_t)b * Tc * BPRc;
  const int NCH = Tc / CHK;                    // 16
  const int NDQ = CHK * BPRc / 16;             // 1040 x b128 per chunk

  { // stage last chunk
    const unsigned char* src = bpb + (size_t)(NCH - 1) * CHK * BPRc;
    const unsigned base = lds_off(&bpl[(NCH - 1) & 1][0]);
    for (int i = tid; i < NDQ; i += 256) async_ld_b128(base + i * 16, i * 16, src);
  }
  async_wait0();
  __syncthreads();

  int wi = (tid == 0) ? sh_wi : 0;
  for (int c = NCH - 1; c >= 0; --c) {
    if (c > 0) { // prefetch the earlier-in-time chunk into the other buffer
      const unsigned char* src = bpb + (size_t)(c - 1) * CHK * BPRc;
      const unsigned base = lds_off(&bpl[(c - 1) & 1][0]);
      for (int i = tid; i < NDQ; i += 256) async_ld_b128(base + i * 16, i * 16, src);
    }
    if (tid == 0) {
      const unsigned char* Bl = bpl[c & 1];
      const int t0 = c * CHK;
      for (int tt = CHK - 1; tt >= 0; --tt) {
        wis[b * Tc + t0 + tt] = wi;            // record BEFORE following bp (matches reverse scan)
        wi -= (int)Bl[tt * BPRc + wi];         // bp[t][wi] = wi - delta  (t=0 result unused)
      }
    }
    async_wait0();
    __syncthreads();
  }
}

// ---------------- Kernel 4: one-hot align output ----------------
// grid = B*T blocks, 288 threads (9 waves, covers U=257)
__global__ __launch_bounds__(288) void ctc_align(const int* __restrict__ wis,
                                                 float* __restrict__ out) {
  const int bt = blockIdx.x;
  const int u  = threadIdx.x;
  const int w  = wis[bt];
  if (u < Uc) out[(size_t)bt * Uc + u] = (u == w) ? 1.0f : 0.0f;
}

// ---------------- host-side launch ----------------
extern "C" void kernel_launch(void* const* d_in, const int* in_sizes, int n_in,
                              void* d_out, int out_size, void* d_ws, size_t ws_size,
                              hipStream_t stream) {
  (void)in_sizes; (void)n_in; (void)out_size; (void)ws_size;
  const float* logits = (const float*)d_in[0];   // [B,T,V] f32
  const int*   label  = (const int*)d_in[1];     // [B,U0] int
  float* out = (float*)d_out;                    // align [B,T,U] then loss [B]

  char* ws = (char*)d_ws;
  const size_t emitBytes = (size_t)Bc * Tc * ERc * sizeof(float);   // 17,301,504
  const size_t bpBytes   = (size_t)Bc * Tc * BPRc;                  //  8,519,680
  const size_t dlBytes   = (size_t)Bc * DLS * sizeof(float);        //     33,792
  float*         emitC  = (float*)ws;
  unsigned char* bpbuf  = (unsigned char*)(ws + emitBytes);
  float*         dpLast = (float*)(ws + emitBytes + bpBytes);
  int*           wisbuf = (int*)(ws + emitBytes + bpBytes + dlBytes);
  float*         outLoss = out + (size_t)Bc * Tc * Uc;              // 8,421,376

  ctc_lsm  <<<Bc * Tc, 256, 0, stream>>>(logits, label, emitC);
  ctc_dp   <<<Bc,      320, 0, stream>>>(emitC, label, bpbuf, dpLast);
  ctc_back <<<Bc,      256, 0, stream>>>(bpbuf, dpLast, label, wisbuf, outLoss);
  ctc_align<<<Bc * Tc, 288, 0, stream>>>(wisbuf, out);
}